// HeteroGraphPolicyNetwork_37649683317224
// MI455X (gfx1250) — compile-verified
//
#include <hip/hip_runtime.h>
#include <math.h>

typedef float v2f __attribute__((ext_vector_type(2)));
typedef float v8f __attribute__((ext_vector_type(8)));

// ---------------- problem constants (match reference) ----------------
#define BG    1000
#define TRI_N 1000
#define SEG_N 1500
#define PT_N  500
#define ANG_N 3000
#define NT_N  (BG*TRI_N)    // 1,000,000 triangles
#define NS_N  (BG*SEG_N)    // 1,500,000 segments
#define NPT_N (BG*PT_N)     //   500,000 points
#define NA_N  (BG*ANG_N)    // 3,000,000 angles

// ---------------- edge scatter kernels (segment_sum via atomics) ----------------
__global__ void hg_scatter_ones_k(const int* __restrict__ dst, float* __restrict__ deg, int E) {
  int e = blockIdx.x * blockDim.x + threadIdx.x;
  if (e < E) atomicAdd(&deg[dst[e]], 1.0f);
}

// seg_f = [boundary, segment_type]
__global__ void hg_scatter_seg2_k(const int* __restrict__ src, const int* __restrict__ dst,
                                  const float* __restrict__ bnd, const float* __restrict__ st,
                                  float* __restrict__ agg, int E) {
  int e = blockIdx.x * blockDim.x + threadIdx.x;
  if (e >= E) return;
  int s = src[e], d = dst[e];
  atomicAdd(&agg[(size_t)2*d + 0], bnd[s]);
  atomicAdd(&agg[(size_t)2*d + 1], st[s]);
}

// ang_f = [angle_type(4), light_cone_angle]
__global__ void hg_scatter_ang5_k(const int* __restrict__ src, const int* __restrict__ dst,
                                  const float* __restrict__ at, const float* __restrict__ lca,
                                  float* __restrict__ agg, int E) {
  int e = blockIdx.x * blockDim.x + threadIdx.x;
  if (e >= E) return;
  int s = src[e], d = dst[e];
  atomicAdd(&agg[(size_t)5*d + 0], at[(size_t)4*s + 0]);
  atomicAdd(&agg[(size_t)5*d + 1], at[(size_t)4*s + 1]);
  atomicAdd(&agg[(size_t)5*d + 2], at[(size_t)4*s + 2]);
  atomicAdd(&agg[(size_t)5*d + 3], at[(size_t)4*s + 3]);
  atomicAdd(&agg[(size_t)5*d + 4], lca[s]);
}

__global__ void hg_scatter_f1_k(const int* __restrict__ src, const int* __restrict__ dst,
                                const float* __restrict__ f, float* __restrict__ agg, int E) {
  int e = blockIdx.x * blockDim.x + threadIdx.x;
  if (e >= E) return;
  atomicAdd(&agg[dst[e]], f[src[e]]);
}

// one thread per (edge, channel): coalesced row reads, contiguous atomics
template <int D>
__global__ void hg_scatter_row_k(const int* __restrict__ src, const int* __restrict__ dst,
                                 const float* __restrict__ H, float* __restrict__ agg, int E) {
  int idx = blockIdx.x * blockDim.x + threadIdx.x;
  int e = idx / D;
  int c = idx % D;
  if (e >= E) return;
  atomicAdd(&agg[(size_t)dst[e]*D + c], H[(size_t)src[e]*D + c]);
}

// ---------------- layer-1 node kernels (tiny d_in: scalar VALU) ----------------
__global__ void hg_tri1_k(const float* __restrict__ tri_t,
                          const float* __restrict__ agg, const float* __restrict__ deg,
                          const float* __restrict__ Wn, const float* __restrict__ Ws,
                          const float* __restrict__ b, float* __restrict__ out, int N) {
  int i = blockIdx.x * blockDim.x + threadIdx.x;
  if (i >= N) return;
  float td  = tri_t[i];
  float inv = 1.0f / fmaxf(deg[i], 1.0f);
  float h0 = agg[(size_t)2*i + 0] * inv;
  float h1 = agg[(size_t)2*i + 1] * inv;
#pragma unroll
  for (int n = 0; n < 16; ++n)
    out[(size_t)i*16 + n] = tanhf(b[n] + td*Ws[n] + h0*Wn[n] + h1*Wn[16 + n]);
}

__global__ void hg_ang1_k(const float* __restrict__ at, const float* __restrict__ lca,
                          const float* __restrict__ agg_sa, const float* __restrict__ deg_sa,
                          const float* __restrict__ agg_ta, const float* __restrict__ deg_ta,
                          const float* __restrict__ Wn_sa, const float* __restrict__ Ws_sa, const float* __restrict__ b_sa,
                          const float* __restrict__ Wn_ta, const float* __restrict__ Ws_ta, const float* __restrict__ b_ta,
                          float* __restrict__ out, int N) {
  int i = blockIdx.x * blockDim.x + threadIdx.x;
  if (i >= N) return;
  float f[5];
#pragma unroll
  for (int k = 0; k < 4; ++k) f[k] = at[(size_t)4*i + k];
  f[4] = lca[i];
  float inva = 1.0f / fmaxf(deg_sa[i], 1.0f);
  float h0 = agg_sa[(size_t)2*i + 0] * inva;
  float h1 = agg_sa[(size_t)2*i + 1] * inva;
  float invt = 1.0f / fmaxf(deg_ta[i], 1.0f);
  float ht = agg_ta[i] * invt;
#pragma unroll
  for (int n = 0; n < 16; ++n) {
    float u = b_sa[n] + h0*Wn_sa[n] + h1*Wn_sa[16 + n];
    float v = b_ta[n] + ht*Wn_ta[n];
#pragma unroll
    for (int k = 0; k < 5; ++k) {
      u += f[k] * Ws_sa[k*16 + n];
      v += f[k] * Ws_ta[k*16 + n];
    }
    out[(size_t)i*16 + n] = 0.5f * (tanhf(u) + tanhf(v));
  }
}

__global__ void hg_pt1_k(const float* __restrict__ nat, const float* __restrict__ nlca,
                         const float* __restrict__ agg_sp, const float* __restrict__ deg_sp,
                         const float* __restrict__ agg_ap, const float* __restrict__ deg_ap,
                         const float* __restrict__ Wn_sp, const float* __restrict__ Ws_sp, const float* __restrict__ b_sp,
                         const float* __restrict__ Wn_ap, const float* __restrict__ Ws_ap, const float* __restrict__ b_ap,
                         float* __restrict__ out, int N) {
  int i = blockIdx.x * blockDim.x + threadIdx.x;
  if (i >= N) return;
  float f[5];
#pragma unroll
  for (int k = 0; k < 4; ++k) f[k] = nat[(size_t)4*i + k];
  f[4] = nlca[i];
  float isp = 1.0f / fmaxf(deg_sp[i], 1.0f);
  float s0 = agg_sp[(size_t)2*i + 0] * isp;
  float s1 = agg_sp[(size_t)2*i + 1] * isp;
  float iap = 1.0f / fmaxf(deg_ap[i], 1.0f);
  float ha[5];
#pragma unroll
  for (int k = 0; k < 5; ++k) ha[k] = agg_ap[(size_t)5*i + k] * iap;
#pragma unroll
  for (int n = 0; n < 16; ++n) {
    float u = b_sp[n] + s0*Wn_sp[n] + s1*Wn_sp[16 + n];
    float v = b_ap[n];
#pragma unroll
    for (int k = 0; k < 5; ++k) {
      u += f[k] * Ws_sp[k*16 + n];
      v += f[k] * Ws_ap[k*16 + n] + ha[k] * Wn_ap[k*16 + n];
    }
    out[(size_t)i*16 + n] = 0.5f * (tanhf(u) + tanhf(v));
  }
}

// ---------------- layer-2: WMMA f32 16x16x4 (exact f32 math) ----------------
// out(16 nodes x 8) = tanh( Hdst(16x16) @ Ws(16x8) + (Agg/deg)(16x16) @ Wn(16x8) + b )
// One wave32 per 16-node tile. Weights staged into LDS zero-padded to 16x16 so the
// B-operand fetch is branch-free (no EXEC save/restore in the WMMA inner loop);
// EXEC stays all-1s for every wave that reaches the WMMA ops (tail skip is wave-uniform).
// A layout (16x4 f32): lanes 0-15 hold M=lane {K0,K1}; lanes 16-31 hold M=lane-16 {K2,K3}.
// B layout (4x16 f32): vgpr0 = row K0 (lo lanes) / K2 (hi lanes); vgpr1 = K1 / K3; N = lane&15.
// C/D layout (16x16 f32): lane holds column N=lane&15, rows M = 8*(lane>>4) + j (j = vgpr 0..7).
__global__ void hg_sage_l2_wmma_k(const float* __restrict__ Hdst,   // N x 16
                                  const float* __restrict__ Agg,    // N x 16
                                  const float* __restrict__ Deg,    // N
                                  const float* __restrict__ Ws,     // 16 x 8
                                  const float* __restrict__ Wn,     // 16 x 8
                                  const float* __restrict__ Bias,   // 8
                                  float* __restrict__ Out,          // N x 8
                                  int nTiles) {
  __shared__ float sWs[16 * 16];   // zero-padded: row k, col n (n>=8 -> 0)
  __shared__ float sWn[16 * 16];
  __shared__ float sB[16];

  {
    int t = threadIdx.x;           // blockDim.x == 256 == 16*16
    int wrow = t >> 4, wcol = t & 15;
    bool v = (wcol < 8);
    sWs[t] = v ? Ws[wrow * 8 + wcol] : 0.0f;
    sWn[t] = v ? Wn[wrow * 8 + wcol] : 0.0f;
    if (t < 16) sB[t] = (t < 8) ? Bias[t] : 0.0f;
  }
  __syncthreads();

  int gtid = blockIdx.x * blockDim.x + threadIdx.x;
  int wave = gtid >> 5;
  int lane = gtid & 31;
  if (wave >= nTiles) return;           // wave-uniform
  const int base = wave << 4;
  const int hi   = lane >> 4;           // selects K pair {0,1} vs {2,3} of each step
  const int r    = lane & 15;           // node row for A, output column for B/D
  const int row  = base + r;
  const float invdeg = 1.0f / fmaxf(Deg[row], 1.0f);

  v8f acc;
  {
    float bval = sB[r];
#pragma unroll
    for (int j = 0; j < 8; ++j) acc[j] = bval;
  }

  const float* hrow = Hdst + (size_t)row * 16;
  const float* arow = Agg  + (size_t)row * 16;

#pragma unroll
  for (int k = 0; k < 4; ++k) {         // h_dst @ Ws
    int k0 = 4*k + 2*hi;
    v2f a; a.x = hrow[k0]; a.y = hrow[k0 + 1];
    v2f b; b.x = sWs[(k0    ) * 16 + r];
           b.y = sWs[(k0 + 1) * 16 + r];
    acc = __builtin_amdgcn_wmma_f32_16x16x4_f32(false, a, false, b, (short)0, acc, false, false);
  }
#pragma unroll
  for (int k = 0; k < 4; ++k) {         // h_neigh @ Wn
    int k0 = 4*k + 2*hi;
    v2f a; a.x = arow[k0] * invdeg; a.y = arow[k0 + 1] * invdeg;
    v2f b; b.x = sWn[(k0    ) * 16 + r];
           b.y = sWn[(k0 + 1) * 16 + r];
    acc = __builtin_amdgcn_wmma_f32_16x16x4_f32(false, a, false, b, (short)0, acc, false, false);
  }

  if (r < 8) {                          // only 8 output columns are real
#pragma unroll
    for (int j = 0; j < 8; ++j) {
      int m = (hi << 3) + j;
      Out[(size_t)(base + m) * 8 + r] = tanhf(acc[j]);
    }
  }
}

// ---------------- layer-3: point logits (8 -> 1) ----------------
__global__ void hg_l3_k(const float* __restrict__ pt2, const float* __restrict__ agg,
                        const float* __restrict__ deg,
                        const float* __restrict__ Ws, const float* __restrict__ Wn,
                        const float* __restrict__ b, float* __restrict__ out, int N) {
  int i = blockIdx.x * blockDim.x + threadIdx.x;
  if (i >= N) return;
  float inv = 1.0f / fmaxf(deg[i], 1.0f);
  float acc = b[0];
#pragma unroll
  for (int j = 0; j < 8; ++j)
    acc += pt2[(size_t)i*8 + j] * Ws[j] + agg[(size_t)i*8 + j] * inv * Wn[j];
  out[i] = acc;
}

// ---------------- graph feature reduction + MLP ----------------
__device__ __forceinline__ float hg_block_sum(float v, float* sd) {
  int t = threadIdx.x;
  sd[t] = v; __syncthreads();
  for (int s = 128; s > 0; s >>= 1) {
    if (t < s) sd[t] += sd[t + s];
    __syncthreads();
  }
  float r = sd[0]; __syncthreads();
  return r;
}

__global__ void hg_gfeat_k(const float* __restrict__ tri_t, const float* __restrict__ seg_t,
                           const float* __restrict__ bnd, const float* __restrict__ nlca,
                           const float* __restrict__ nat, float* __restrict__ gfeat) {
  __shared__ float sd[256];
  int b = blockIdx.x, t = threadIdx.x;
  float s_tri = 0.0f;
  for (int i = t; i < TRI_N; i += 256) s_tri += tri_t[(size_t)b*TRI_N + i];
  float s_s = 0.0f, s_b = 0.0f, s_sb = 0.0f;
  for (int i = t; i < SEG_N; i += 256) {
    float s = seg_t[(size_t)b*SEG_N + i];
    float bb = bnd[(size_t)b*SEG_N + i];
    s_s += s; s_b += bb; s_sb += s * bb;
  }
  float s_l = 0.0f, s_a[4] = {0.0f, 0.0f, 0.0f, 0.0f};
  for (int i = t; i < PT_N; i += 256) {
    size_t idx = (size_t)b*PT_N + i;
    s_l += nlca[idx];
#pragma unroll
    for (int k = 0; k < 4; ++k) s_a[k] += nat[idx*4 + k];
  }
  s_tri = hg_block_sum(s_tri, sd);
  s_s   = hg_block_sum(s_s, sd);
  s_b   = hg_block_sum(s_b, sd);
  s_sb  = hg_block_sum(s_sb, sd);
  s_l   = hg_block_sum(s_l, sd);
  for (int k = 0; k < 4; ++k) s_a[k] = hg_block_sum(s_a[k], sd);
  if (t == 0) {
    float* g = gfeat + (size_t)b * 15;
    g[0] = 6.90775527898214f;   // ln(1000)
    g[1] = 7.31322038709030f;   // ln(1500)
    g[2] = 6.21460809842219f;   // ln(500)
    float mt = s_tri * (1.0f / TRI_N); g[3] = 1.0f - mt; g[4] = mt;
    float ms = s_s   * (1.0f / SEG_N); g[5] = 1.0f - ms; g[6] = ms;
    g[7] = s_b  * (1.0f / SEG_N);
    g[8] = (s_b - s_sb) * (1.0f / SEG_N);
    g[9] = s_sb * (1.0f / SEG_N);
    g[10] = s_l * (1.0f / (4.0f * PT_N));
    for (int k = 0; k < 4; ++k) g[11 + k] = s_a[k] * (1.0f / PT_N);
  }
}

__global__ void hg_gmlp_k(const float* __restrict__ gfeat,
                          const float* __restrict__ W1, const float* __restrict__ b1,
                          const float* __restrict__ W2, const float* __restrict__ b2,
                          const float* __restrict__ W3, const float* __restrict__ b3,
                          float* __restrict__ out) {
  int g = blockIdx.x * blockDim.x + threadIdx.x;
  if (g >= BG) return;
  float x[15];
#pragma unroll
  for (int j = 0; j < 15; ++j) x[j] = gfeat[(size_t)g*15 + j];
  float h[32];
  for (int n = 0; n < 32; ++n) {
    float a = b1[n];
#pragma unroll
    for (int j = 0; j < 15; ++j) a += x[j] * W1[j*32 + n];
    h[n] = tanhf(a);
  }
  float h2[16];
  for (int n = 0; n < 16; ++n) {
    float a = b2[n];
#pragma unroll
    for (int j = 0; j < 32; ++j) a += h[j] * W2[j*16 + n];
    h2[n] = tanhf(a);
  }
  for (int n = 0; n < 7; ++n) {
    float a = b3[n];
#pragma unroll
    for (int j = 0; j < 16; ++j) a += h2[j] * W3[j*7 + n];
    out[(size_t)g*7 + n] = a;
  }
}

// ---------------- launch ----------------
extern "C" void kernel_launch(void* const* d_in, const int* in_sizes, int n_in,
                              void* d_out, int out_size, void* d_ws, size_t ws_size,
                              hipStream_t stream) {
  (void)n_in; (void)out_size; (void)ws_size;
  // node features
  const float* tri_t = (const float*)d_in[0];
  const float* seg_t = (const float*)d_in[1];
  const float* bnd   = (const float*)d_in[2];
  const float* at    = (const float*)d_in[3];
  const float* lca   = (const float*)d_in[4];
  const float* nat   = (const float*)d_in[5];
  const float* nlca  = (const float*)d_in[6];
  // edges
  const int* st_src = (const int*)d_in[7];  const int* st_dst = (const int*)d_in[8];
  const int* sp_src = (const int*)d_in[9];  const int* sp_dst = (const int*)d_in[10];
  const int* sa_src = (const int*)d_in[11]; const int* sa_dst = (const int*)d_in[12];
  const int* ap_src = (const int*)d_in[13]; const int* ap_dst = (const int*)d_in[14];
  const int* ta_src = (const int*)d_in[15]; const int* ta_dst = (const int*)d_in[16];
  const int eST = in_sizes[7], eSP = in_sizes[9], eSA = in_sizes[11];
  const int eAP = in_sizes[13], eTA = in_sizes[15];
  // params pytree leaves (JAX flatten order: dict keys sorted; 'W*' < 'b*')
  const float* gW1 = (const float*)d_in[17];
  const float* gW2 = (const float*)d_in[18];
  const float* gW3 = (const float*)d_in[19];
  const float* gb1 = (const float*)d_in[20];
  const float* gb2 = (const float*)d_in[21];
  const float* gb3 = (const float*)d_in[22];
  const float* ap1_Wn = (const float*)d_in[23]; const float* ap1_Ws = (const float*)d_in[24]; const float* ap1_b = (const float*)d_in[25];
  const float* sa1_Wn = (const float*)d_in[26]; const float* sa1_Ws = (const float*)d_in[27]; const float* sa1_b = (const float*)d_in[28];
  const float* sp1_Wn = (const float*)d_in[29]; const float* sp1_Ws = (const float*)d_in[30]; const float* sp1_b = (const float*)d_in[31];
  const float* st1_Wn = (const float*)d_in[32]; const float* st1_Ws = (const float*)d_in[33]; const float* st1_b = (const float*)d_in[34];
  const float* ta1_Wn = (const float*)d_in[35]; const float* ta1_Ws = (const float*)d_in[36]; const float* ta1_b = (const float*)d_in[37];
  const float* ap2_Wn = (const float*)d_in[38]; const float* ap2_Ws = (const float*)d_in[39]; const float* ap2_b = (const float*)d_in[40];
  const float* ta2_Wn = (const float*)d_in[41]; const float* ta2_Ws = (const float*)d_in[42]; const float* ta2_b = (const float*)d_in[43];
  const float* ap3_Wn = (const float*)d_in[44]; const float* ap3_Ws = (const float*)d_in[45]; const float* ap3_b = (const float*)d_in[46];

  // ---- workspace layout (floats) ----
  float* ws = (float*)d_ws;
  size_t off = 0;
  float* deg_st = ws + off; off += NT_N;
  float* deg_sp = ws + off; off += NPT_N;
  float* deg_sa = ws + off; off += NA_N;
  float* deg_ap = ws + off; off += NPT_N;
  float* deg_ta = ws + off; off += NA_N;
  const size_t degTotal = off;
  float* tri1  = ws + off; off += (size_t)NT_N * 16;
  float* pt1   = ws + off; off += (size_t)NPT_N * 16;
  float* ang1  = ws + off; off += (size_t)NA_N * 16;
  float* pt2   = ws + off; off += (size_t)NPT_N * 8;
  float* ang2  = ws + off; off += (size_t)NA_N * 8;
  float* gfeat = ws + off; off += (size_t)BG * 15;
  float* arena = ws + off;
  // layer-1 aggregation aliases (arena pass 1)
  float* agg_st = arena;
  float* agg_sp = agg_st + (size_t)2 * NT_N;
  float* agg_sa = agg_sp + (size_t)2 * NPT_N;
  float* agg_ap = agg_sa + (size_t)2 * NA_N;
  float* agg_ta = agg_ap + (size_t)5 * NPT_N;
  const size_t agg1Total = (size_t)2*NT_N + 2*NPT_N + 2*NA_N + 5*NPT_N + NA_N;
  // layer-2 aggregation aliases (arena pass 2)
  float* agg2_ap = arena;
  float* agg2_ta = agg2_ap + (size_t)16 * NPT_N;
  const size_t agg2Total = (size_t)16*NPT_N + (size_t)16*NA_N;
  // layer-3 aggregation alias (arena pass 3; pt2 already consumed agg2_ap)
  float* agg3_ap = arena;

  float* out_pts   = (float*)d_out;        // NPT point logits
  float* out_graph = out_pts + NPT_N;      // B*7 graph logits

  const int T = 256;
  auto nb = [](long n) { return (unsigned)((n + 255) / 256); };

  // ---- phase 0: zero accumulators (poisoned workspace; per-launch determinism) ----
  (void)hipMemsetAsync(ws, 0, degTotal * sizeof(float), stream);
  (void)hipMemsetAsync(arena, 0, agg1Total * sizeof(float), stream);

  // ---- phase 1: degrees + layer-1 aggregations ----
  hg_scatter_ones_k<<<nb(eST), T, 0, stream>>>(st_dst, deg_st, eST);
  hg_scatter_ones_k<<<nb(eSP), T, 0, stream>>>(sp_dst, deg_sp, eSP);
  hg_scatter_ones_k<<<nb(eSA), T, 0, stream>>>(sa_dst, deg_sa, eSA);
  hg_scatter_ones_k<<<nb(eAP), T, 0, stream>>>(ap_dst, deg_ap, eAP);
  hg_scatter_ones_k<<<nb(eTA), T, 0, stream>>>(ta_dst, deg_ta, eTA);

  hg_scatter_seg2_k<<<nb(eST), T, 0, stream>>>(st_src, st_dst, bnd, seg_t, agg_st, eST);
  hg_scatter_seg2_k<<<nb(eSP), T, 0, stream>>>(sp_src, sp_dst, bnd, seg_t, agg_sp, eSP);
  hg_scatter_seg2_k<<<nb(eSA), T, 0, stream>>>(sa_src, sa_dst, bnd, seg_t, agg_sa, eSA);
  hg_scatter_ang5_k<<<nb(eAP), T, 0, stream>>>(ap_src, ap_dst, at, lca, agg_ap, eAP);
  hg_scatter_f1_k<<<nb(eTA), T, 0, stream>>>(ta_src, ta_dst, tri_t, agg_ta, eTA);

  // ---- layer-1 node transforms ----
  hg_tri1_k<<<nb(NT_N), T, 0, stream>>>(tri_t, agg_st, deg_st, st1_Wn, st1_Ws, st1_b, tri1, NT_N);
  hg_ang1_k<<<nb(NA_N), T, 0, stream>>>(at, lca, agg_sa, deg_sa, agg_ta, deg_ta,
                                        sa1_Wn, sa1_Ws, sa1_b, ta1_Wn, ta1_Ws, ta1_b, ang1, NA_N);
  hg_pt1_k<<<nb(NPT_N), T, 0, stream>>>(nat, nlca, agg_sp, deg_sp, agg_ap, deg_ap,
                                        sp1_Wn, sp1_Ws, sp1_b, ap1_Wn, ap1_Ws, ap1_b, pt1, NPT_N);

  // ---- layer-2: aggregate 16-wide hidden states, then WMMA transforms ----
  (void)hipMemsetAsync(arena, 0, agg2Total * sizeof(float), stream);
  hg_scatter_row_k<16><<<nb((long)eAP * 16), T, 0, stream>>>(ap_src, ap_dst, ang1, agg2_ap, eAP);
  hg_scatter_row_k<16><<<nb((long)eTA * 16), T, 0, stream>>>(ta_src, ta_dst, tri1, agg2_ta, eTA);

  const int tilesP = NPT_N / 16, tilesA = NA_N / 16;
  hg_sage_l2_wmma_k<<<nb((long)tilesP * 32), T, 0, stream>>>(pt1, agg2_ap, deg_ap,
                                                             ap2_Ws, ap2_Wn, ap2_b, pt2, tilesP);
  hg_sage_l2_wmma_k<<<nb((long)tilesA * 32), T, 0, stream>>>(ang1, agg2_ta, deg_ta,
                                                             ta2_Ws, ta2_Wn, ta2_b, ang2, tilesA);

  // ---- layer-3: aggregate 8-wide, point logits ----
  (void)hipMemsetAsync(agg3_ap, 0, (size_t)8 * NPT_N * sizeof(float), stream);
  hg_scatter_row_k<8><<<nb((long)eAP * 8), T, 0, stream>>>(ap_src, ap_dst, ang2, agg3_ap, eAP);
  hg_l3_k<<<nb(NPT_N), T, 0, stream>>>(pt2, agg3_ap, deg_ap, ap3_Ws, ap3_Wn, ap3_b, out_pts, NPT_N);

  // ---- graph head ----
  hg_gfeat_k<<<BG, T, 0, stream>>>(tri_t, seg_t, bnd, nlca, nat, gfeat);
  hg_gmlp_k<<<nb(BG), T, 0, stream>>>(gfeat, gW1, gb1, gW2, gb2, gW3, gb3, out_graph);
}